// lowLightLoss_4810363373040
// MI455X (gfx1250) — compile-verified
//
#include <hip/hip_runtime.h>
#include <math.h>

// ---------------------------------------------------------------------------
// Low-light Retinex-style loss, MI455X (gfx1250, wave32).
// Memory-bound streaming problem: ~300MB total traffic -> ~13us @ 23.3TB/s.
// WMMA (V_WMMA_F32_16X16X4_F32) is used as an exact f32 cross-lane reducer.
// ---------------------------------------------------------------------------

#define HH  2048
#define WW  2048
#define HWN (HH * WW)              // 4,194,304 pixels per plane
#define NP4 ((HH + 1) * WW)        // gradient grid rows 0..H

typedef float v2f __attribute__((ext_vector_type(2)));
typedef float v8f __attribute__((ext_vector_type(8)));

// ---------------------------------------------------------------------------
// Exact 32-lane f32 sum via one V_WMMA_F32_16X16X4_F32.
// A-matrix (16x4, 2 VGPRs): lane m (m<16) supplies A[m,0]=v (VGPR0), A[m,1]=0;
// lane m+16 supplies A[m,2]=v, A[m,3]=0.  B = all ones (layout-agnostic).
// D[m,n] = v_m + v_{m+16} for every n.  Summing a lane's 8 D VGPRs gives the
// sum over M=0..7 (lanes 0-15) or M=8..15 (lanes 16-31); one xor-16 shuffle
// combines the two halves into the full wave sum on every lane.
// ---------------------------------------------------------------------------
__device__ __forceinline__ float wave_sum(float v) {
#if defined(__has_builtin) && __has_builtin(__builtin_amdgcn_wmma_f32_16x16x4_f32)
    v2f a; a[0] = v;    a[1] = 0.0f;
    v2f b; b[0] = 1.0f; b[1] = 1.0f;
    v8f c = {0.0f, 0.0f, 0.0f, 0.0f, 0.0f, 0.0f, 0.0f, 0.0f};
    c = __builtin_amdgcn_wmma_f32_16x16x4_f32(false, a, false, b,
                                              (short)0, c, false, false);
    float t = c[0] + c[1] + c[2] + c[3] + c[4] + c[5] + c[6] + c[7];
    t += __shfl_xor(t, 16, 32);
    return t;
#else
    for (int off = 16; off >= 1; off >>= 1) v += __shfl_xor(v, off, 32);
    return v;
#endif
}

// Block reduce (<=8 wave32 waves) + one f64 atomic per block.
__device__ __forceinline__ void block_sum_atomic(float v, double* target,
                                                 float* smem8) {
    float w = wave_sum(v);
    const int lane = threadIdx.x & 31;
    const int wid  = threadIdx.x >> 5;
    if (lane == 0) smem8[wid] = w;
    __syncthreads();
    if (threadIdx.x == 0) {
        float s = 0.0f;
        const int nw = blockDim.x >> 5;
        for (int i = 0; i < nw; ++i) s += smem8[i];
        atomicAdd(target, (double)s);
    }
    __syncthreads();  // allow smem8 reuse by the next call
}

// ---------------------------------------------------------------------------
// Workspace layout (total ~4.2 MB)
// ---------------------------------------------------------------------------
// [0, HWN)                : Rg grayscale plane, uint8 (exact: Rg = k/255)
// [HWN, +1024)            : histogram, 256 x u32
// [HWN+1024, +1024)       : cdf, 256 x f32
// [HWN+2048, +8)          : min/max bits of im_max (u32 each)
// [HWN+2064, +32)         : sums[4] f64: recon_low, recon_eq, r_smooth, ismooth

__global__ void k_init(unsigned* hist, unsigned* minmax, double* sums) {
    const int t = threadIdx.x;
    if (t < 256) hist[t] = 0u;
    if (t == 0) {
        minmax[0] = 0x7F800000u;  // +inf  (im_max >= 0, so bit-order == order)
        minmax[1] = 0x00000000u;  // +0
        sums[0] = 0.0; sums[1] = 0.0; sums[2] = 0.0; sums[3] = 0.0;
    }
}

// Pass 1: stream all inputs once. Produces Rg (u8), recon_low sum, min/max.
__global__ void k_pass1(const float* __restrict__ im, const float* __restrict__ R,
                        const float* __restrict__ L, unsigned char* __restrict__ rg,
                        unsigned* minmax, double* sums) {
    __shared__ float sm_sum[8], sm_min[8], sm_max[8];
    float acc = 0.0f, mn = INFINITY, mx = 0.0f;
    const int stride = gridDim.x * blockDim.x;
    for (int p = blockIdx.x * blockDim.x + threadIdx.x; p < HWN; p += stride) {
        const float i0 = im[p], i1 = im[p + HWN], i2 = im[p + 2 * HWN];
        const float r0 = R[p],  r1 = R[p + HWN],  r2 = R[p + 2 * HWN];
        const float l  = L[p];
        const float imx = fmaxf(i0, fmaxf(i1, i2));
        mn = fminf(mn, imx);
        mx = fmaxf(mx, imx);
        acc += fabsf(fmaf(r0, l, -i0)) + fabsf(fmaf(r1, l, -i1))
             + fabsf(fmaf(r2, l, -i2));
        // PIL 'L' grayscale: q=floor(clip(r)*255); k=(q0*19595+q1*38470+q2*7471+32768)>>16
        // all intermediates are exact integers < 2^24 in f32.
        const float q0 = floorf(fminf(fmaxf(r0, 0.0f), 1.0f) * 255.0f);
        const float q1 = floorf(fminf(fmaxf(r1, 0.0f), 1.0f) * 255.0f);
        const float q2 = floorf(fminf(fmaxf(r2, 0.0f), 1.0f) * 255.0f);
        const float s  = q0 * 19595.0f + q1 * 38470.0f + q2 * 7471.0f + 32768.0f;
        rg[p] = (unsigned char)floorf(s * (1.0f / 65536.0f));
    }
    // min / max: shuffle tree, then bit-atomics (values are non-negative)
    for (int off = 16; off >= 1; off >>= 1) {
        mn = fminf(mn, __shfl_xor(mn, off, 32));
        mx = fmaxf(mx, __shfl_xor(mx, off, 32));
    }
    const int lane = threadIdx.x & 31, wid = threadIdx.x >> 5;
    if (lane == 0) { sm_min[wid] = mn; sm_max[wid] = mx; }
    __syncthreads();
    if (threadIdx.x == 0) {
        float bm = sm_min[0], bx = sm_max[0];
        const int nw = blockDim.x >> 5;
        for (int i = 1; i < nw; ++i) {
            bm = fminf(bm, sm_min[i]);
            bx = fmaxf(bx, sm_max[i]);
        }
        atomicMin(&minmax[0], __float_as_uint(bm));
        atomicMax(&minmax[1], __float_as_uint(bx));
    }
    block_sum_atomic(acc, &sums[0], sm_sum);
}

// Pass 2: histogram of im_max (recomputed from input; hits L2).
__global__ void k_pass2(const float* __restrict__ im, const unsigned* minmax,
                        unsigned* hist) {
    __shared__ unsigned hs[256];
    for (int k = threadIdx.x; k < 256; k += blockDim.x) hs[k] = 0u;
    __syncthreads();
    const float mn = __uint_as_float(minmax[0]);
    const float mx = __uint_as_float(minmax[1]);
    const float range = mx - mn;
    const float scale = (range > 0.0f) ? (256.0f / range) : 0.0f;
    const int stride = gridDim.x * blockDim.x;
    for (int p = blockIdx.x * blockDim.x + threadIdx.x; p < HWN; p += stride) {
        const float v = fmaxf(im[p], fmaxf(im[p + HWN], im[p + 2 * HWN]));
        int b = (int)((v - mn) * scale);
        b = (b < 0) ? 0 : ((b > 255) ? 255 : b);
        atomicAdd(&hs[b], 1u);
    }
    __syncthreads();
    for (int k = threadIdx.x; k < 256; k += blockDim.x)
        if (hs[k]) atomicAdd(&hist[k], hs[k]);
}

// Pass 3: CDF (256 threads, one block). cdf[k] = cumcount[k] / N.
__global__ void k_pass3(const unsigned* __restrict__ hist, float* __restrict__ cdf) {
    const int k = threadIdx.x;
    unsigned c = 0;
    for (int i = 0; i <= k; ++i) c += hist[i];
    __shared__ unsigned tot;
    if (k == 255) tot = c;
    __syncthreads();
    cdf[k] = (float)((double)c / (double)tot);
}

// Pass 4: interp (recon_eq) + gradient/exp sums over rows 0..H inclusive.
//  a-grid value at (i,j):   i==0 -> Rg[0,j]; i==H -> Rg[H-1,j]; else |dRg|
//  per-pixel (i<H) terms:   2*Rg (r_smooth), 2*|L|*exp(-10*Rg) (ismooth),
//                           |R_max - interp(im_max)| (recon_eq)
__global__ void k_pass4(const float* __restrict__ im, const float* __restrict__ R,
                        const float* __restrict__ L,
                        const unsigned char* __restrict__ rg,
                        const unsigned* minmax, const float* __restrict__ cdf,
                        double* sums) {
    __shared__ float cdfs[256];
    __shared__ float sm[8];
    for (int k = threadIdx.x; k < 256; k += blockDim.x) cdfs[k] = cdf[k];
    __syncthreads();
    const float mn = __uint_as_float(minmax[0]);
    const float mx = __uint_as_float(minmax[1]);
    const float wd = (mx - mn) * (1.0f / 256.0f);
    const float invwd = (wd > 0.0f) ? (1.0f / wd) : 0.0f;

    float e0 = 0.0f, e1 = 0.0f, e2 = 0.0f;  // recon_eq, r_smooth, ismooth
    const int stride = gridDim.x * blockDim.x;
    for (int idx = blockIdx.x * blockDim.x + threadIdx.x; idx < NP4; idx += stride) {
        const int i = idx >> 11;          // / WW (2048)
        const int j = idx & (WW - 1);
        if (i < HH) {
            const int p = idx;
            const float v = fmaxf(im[p], fmaxf(im[p + HWN], im[p + 2 * HWN]));
            const float rmx = fmaxf(R[p], fmaxf(R[p + HWN], R[p + 2 * HWN]));
            // jnp.interp(v, edges[:-1], cdf): clamp outside [edges0, edges255]
            const float u = (v - mn) * invwd;
            int k = (int)floorf(u);
            k = (k < 0) ? 0 : k;
            float ie;
            if (k >= 255) {
                ie = cdfs[255];
            } else {
                const float ek = mn + (float)k * wd;
                ie = cdfs[k] + (cdfs[k + 1] - cdfs[k]) * ((v - ek) * invwd);
            }
            e0 += fabsf(rmx - ie);
            const float g = (float)rg[p] / 255.0f;
            const float l = L[p];
            e1 += 2.0f * g;
            e2 += 2.0f * fabsf(l) * expf(-10.0f * g);
            float aval, dl;
            if (i == 0) {
                aval = g;                 // |Rg[0,j] - 0|
                dl   = fabsf(l);
            } else {
                const float gm = (float)rg[p - WW] / 255.0f;
                aval = fabsf(g - gm);
                dl   = fabsf(l - L[p - WW]);
            }
            e1 += aval;
            e2 += dl * expf(-10.0f * aval);
        } else {                           // i == H bottom boundary row
            const int p = (HH - 1) * WW + j;
            const float gm = (float)rg[p] / 255.0f;
            const float lm = L[p];
            e1 += gm;                      // |0 - Rg[H-1,j]|
            e2 += fabsf(lm) * expf(-10.0f * gm);
        }
    }
    block_sum_atomic(e0, &sums[1], sm);
    block_sum_atomic(e1, &sums[2], sm);
    block_sum_atomic(e2, &sums[3], sm);
}

// Pass 5: combine in f64, emit scalar.
__global__ void k_pass5(const double* __restrict__ sums, float* __restrict__ out) {
    if (threadIdx.x == 0 && blockIdx.x == 0) {
        const double n_recon = 3.0 * (double)HWN;
        const double n_eq    = (double)HWN;
        const double n_grid  = 2.0 * (double)(HH + 1) * (double)(WW + 2);
        const double loss = sums[0] / n_recon
                          + 0.1  * (sums[3] / n_grid)
                          + 0.1  * (sums[1] / n_eq)
                          + 0.01 * (sums[2] / n_grid);
        out[0] = (float)loss;
    }
}

// ---------------------------------------------------------------------------
extern "C" void kernel_launch(void* const* d_in, const int* in_sizes, int n_in,
                              void* d_out, int out_size, void* d_ws, size_t ws_size,
                              hipStream_t stream) {
    (void)in_sizes; (void)n_in; (void)out_size; (void)ws_size;
    const float* im = (const float*)d_in[0];   // [1,3,H,W]
    const float* R  = (const float*)d_in[1];   // [1,3,H,W]
    const float* L  = (const float*)d_in[2];   // [1,1,H,W]
    float* out = (float*)d_out;

    char* w = (char*)d_ws;
    unsigned char* rg  = (unsigned char*)w;
    unsigned* hist     = (unsigned*)(w + HWN);
    float*    cdf      = (float*)   (w + HWN + 1024);
    unsigned* minmax   = (unsigned*)(w + HWN + 2048);
    double*   sums     = (double*)  (w + HWN + 2064);

    k_init <<<dim3(1),    dim3(256), 0, stream>>>(hist, minmax, sums);
    k_pass1<<<dim3(4096), dim3(256), 0, stream>>>(im, R, L, rg, minmax, sums);
    k_pass2<<<dim3(2048), dim3(256), 0, stream>>>(im, minmax, hist);
    k_pass3<<<dim3(1),    dim3(256), 0, stream>>>(hist, cdf);
    k_pass4<<<dim3(4096), dim3(256), 0, stream>>>(im, R, L, rg, minmax, cdf, sums);
    k_pass5<<<dim3(1),    dim3(1),   0, stream>>>(sums, out);
}